// VectorizedLinearCell_8564164788367
// MI455X (gfx1250) — compile-verified
//
#include <hip/hip_runtime.h>
#include <stdint.h>
#include <math.h>

// Linear recurrence scan: z_t = a*x_t + b*z_{t-1} over T, per row.
// B=4096 rows, T=8192, fp32. Memory bound: 268 MB @ 23.3 TB/s ~= 11.5 us floor.
// One 256-thread block per row; each thread owns 32 contiguous floats (one
// 128B cacheline). The row is staged global->LDS by a single TDM
// tensor_load_to_lds (issued by wave 0 before the tanh scalar math so the DMA
// overlaps it), with TDM pad_enable giving a 144B per-thread stride so
// ds_load_b128 hits the minimum possible 2 bank phases. Then a 3-level affine
// scan: serial per-thread, wave32 shuffle scan, cross-wave combine via LDS.

#define T_LEN     8192
#define NTHREADS  256
#define TPT       (T_LEN / NTHREADS)        // 32 floats per thread (one 128B line)
#define NWAVES    (NTHREADS / 32)           // 8 wave32s
#define STRIDE_F  36                        // 32 data dwords + 4 pad dwords (TDM pad)
#define LDS_FLOATS (NTHREADS * STRIDE_F)    // 9216 floats = 36 KB

typedef uint32_t u32;
typedef uint64_t u64;
typedef __attribute__((ext_vector_type(4))) u32 v4u;
typedef __attribute__((ext_vector_type(8))) int v8i;
typedef __attribute__((ext_vector_type(4))) int v4i;

#if defined(__AMDGCN__) && defined(__gfx1250__) && \
    __has_builtin(__builtin_amdgcn_tensor_load_to_lds) && \
    __has_builtin(__builtin_amdgcn_s_wait_tensorcnt)
#define USE_TDM 1
#else
#define USE_TDM 0
#endif

__global__ __launch_bounds__(NTHREADS)
void lrnn_scan_kernel(const float* __restrict__ X,
                      const float* __restrict__ rawa,
                      const float* __restrict__ rawb,
                      float* __restrict__ Z)
{
    const int row  = blockIdx.x;
    const int tid  = threadIdx.x;
    const int lane = tid & 31;
    const int wave = tid >> 5;

    __shared__ float wmS[NWAVES];
    __shared__ float wsS[NWAVES];

#if USE_TDM
    __shared__ float smem[LDS_FLOATS];
    // Issue the TDM DMA FIRST (wave 0 only: EXEC is ignored by tensor ops, so
    // every wave issuing it would multiply traffic). The transfer then runs
    // concurrently with the tanh scalar math below.
    if (tid < 32) {
        u64 gaddr   = (u64)(uintptr_t)(X + (u64)row * T_LEN);
        u32 lds_off = (u32)(uintptr_t)&smem[0];   // low 32 bits of flat shared = LDS offset

        // ---- D# group 0 (ISA 8.3) ----
        v4u g0;
        g0.x = 1u;                                   // count=1 (valid), user mode
        g0.y = lds_off;                              // lds_addr (bytes)
        g0.z = (u32)gaddr;                           // global_addr[31:0]
        g0.w = (u32)((gaddr >> 32) & 0x01FFFFFFu)    // global_addr[56:32]
             | (2u << 30);                           // type = 2 ("image")

        // ---- D# group 1 (ISA 8.4) ----
        u32 w0 = (2u << 16)      // data_size = 4 bytes
               | (1u << 20)      // pad_enable: insert LDS padding
               | (4u << 22)      // pad_interval: 2^(4+1)=32 dwords (128B) between pads
               | (3u << 25);     // pad_amount: 3+1 = 4 dwords (16B) of pad
        u32 w1 = ((u32)(T_LEN & 0xFFFF)) << 16;      // tensor_dim0[15:0]  @ bits 63:48
        u32 w2 = ((u32)(T_LEN >> 16) & 0xFFFFu)      // tensor_dim0[31:16] @ bits 79:64
               | (1u << 16);                         // tensor_dim1 = 1    @ bits 95:80
        u32 w3 = ((u32)T_LEN) << 16;                 // tile_dim0 = 8192   @ bits 127:112
        u32 w4 = 1u;                                 // tile_dim1 = 1, tile_dim2 = 0
        u32 w5 = (u32)T_LEN;                         // tensor_dim0_stride[31:0]
        u32 w6 = 0u;                                 // stride high bits / dim1_stride = 0
        u32 w7 = 0u;
        v8i g1;
        g1[0]=(int)w0; g1[1]=(int)w1; g1[2]=(int)w2; g1[3]=(int)w3;
        g1[4]=(int)w4; g1[5]=(int)w5; g1[6]=(int)w6; g1[7]=(int)w7;
        v4i g2 = {0,0,0,0};
        v4i g3 = {0,0,0,0};
#if __clang_major__ >= 23
        v8i g4 = {0,0,0,0,0,0,0,0};
        __builtin_amdgcn_tensor_load_to_lds(g0, g1, g2, g3, g4, 0);
#else
        __builtin_amdgcn_tensor_load_to_lds(g0, g1, g2, g3, 0);
#endif
    }
#endif

    // Scalar projection params (overlaps the in-flight TDM transfer).
    const float a = 0.5f * tanhf(rawa[0]);
    const float b = tanhf(rawb[0]);

#if USE_TDM
    if (tid < 32) {
        __builtin_amdgcn_s_wait_tensorcnt(0);
    }
    __syncthreads();
    const float* __restrict__ my = &smem[tid * STRIDE_F];
#else
    // Fallback: direct global loads, keep the chunk in registers.
    float xr[TPT];
    {
        const float* rowp = X + (u64)row * T_LEN + (u64)tid * TPT;
        #pragma unroll
        for (int i = 0; i < TPT; i += 4) {
            float4 v = *(const float4*)(rowp + i);
            xr[i+0]=v.x; xr[i+1]=v.y; xr[i+2]=v.z; xr[i+3]=v.w;
        }
    }
#endif

    // ---- pass 1: thread-local serial scan (zero init) -> affine pair (m, s) ----
    float z = 0.0f;
#if USE_TDM
    #pragma unroll
    for (int i = 0; i < TPT; i += 4) {
        float4 v = *(const float4*)(my + i);
        z = fmaf(b, z, a * v.x);
        z = fmaf(b, z, a * v.y);
        z = fmaf(b, z, a * v.z);
        z = fmaf(b, z, a * v.w);
    }
#else
    #pragma unroll
    for (int i = 0; i < TPT; ++i) z = fmaf(b, z, a * xr[i]);
#endif

    float m = b * b; m *= m; m *= m; m *= m; m *= m;   // b^32 (5 squarings)
    float s = z;

    // ---- wave32 inclusive affine scan over lanes (Hillis-Steele) ----
    // combine(earlier=(pm,ps), later=(m,s)) = (pm*m, m*ps + s)
    #pragma unroll
    for (int d = 1; d < 32; d <<= 1) {
        float pm = __shfl_up(m, (unsigned)d, 32);
        float ps = __shfl_up(s, (unsigned)d, 32);
        if (lane >= d) { s = fmaf(m, ps, s); m *= pm; }
    }

    if (lane == 31) { wmS[wave] = m; wsS[wave] = s; }
    __syncthreads();

    // ---- cross-wave exclusive prefix (<= 7 cheap iterations) ----
    float Es = 0.0f;
    for (int j = 0; j < wave; ++j)
        Es = fmaf(wmS[j], Es, wsS[j]);

    // exclusive-within-wave pair (lanes before this one)
    float Lm = __shfl_up(m, 1u, 32);
    float Ls = __shfl_up(s, 1u, 32);
    if (lane == 0) { Lm = 1.0f; Ls = 0.0f; }

    const float h = fmaf(Lm, Es, Ls);   // hidden state entering this thread's chunk

    // ---- pass 2: re-run recurrence seeded with true carry, store 128B/thread ----
    float* outp = Z + (u64)row * T_LEN + (u64)tid * TPT;
    float zz = h;
#if USE_TDM
    #pragma unroll
    for (int i = 0; i < TPT; i += 4) {
        float4 v = *(const float4*)(my + i);
        float4 o;
        zz = fmaf(b, zz, a * v.x); o.x = zz;
        zz = fmaf(b, zz, a * v.y); o.y = zz;
        zz = fmaf(b, zz, a * v.z); o.z = zz;
        zz = fmaf(b, zz, a * v.w); o.w = zz;
        *(float4*)(outp + i) = o;
    }
#else
    #pragma unroll
    for (int i = 0; i < TPT; i += 4) {
        float4 o;
        zz = fmaf(b, zz, a * xr[i+0]); o.x = zz;
        zz = fmaf(b, zz, a * xr[i+1]); o.y = zz;
        zz = fmaf(b, zz, a * xr[i+2]); o.z = zz;
        zz = fmaf(b, zz, a * xr[i+3]); o.w = zz;
        *(float4*)(outp + i) = o;
    }
#endif
}

extern "C" void kernel_launch(void* const* d_in, const int* in_sizes, int n_in,
                              void* d_out, int out_size, void* d_ws, size_t ws_size,
                              hipStream_t stream)
{
    const float* X  = (const float*)d_in[0];   // [B, T] fp32
    const float* ra = (const float*)d_in[1];   // scalar raw_a
    const float* rb = (const float*)d_in[2];   // scalar raw_b
    float* Z = (float*)d_out;                  // [B, T] fp32

    const int B = in_sizes[0] / T_LEN;
    hipLaunchKernelGGL(lrnn_scan_kernel, dim3(B), dim3(NTHREADS), 0, stream,
                       X, ra, rb, Z);
}